// CapsuleNetwork_41154376631002
// MI455X (gfx1250) — compile-verified
//
#include <hip/hip_runtime.h>
#include <math.h>

typedef __attribute__((ext_vector_type(2))) float v2f;
typedef __attribute__((ext_vector_type(8))) float v8f;

#define EPS_SQ 1e-6f
#define EPS_BN 1e-5f
#define KCHUNK 32
#define LDS_STRIDE 34   // 16x32 panel padded: even (b64 align) + bank-spread

// ---------------------------------------------------------------------------
// Implicit-GEMM conv via V_WMMA_F32_16X16X4_F32 (full fp32 matrix pipe).
// One wave computes a 16(M=Cout) x 16(N=B*OH*OW) tile. K = Cin*9 is processed
// in 32-wide chunks staged through LDS with branchless clamped gathers, then
// 8 unrolled WMMAs per chunk read their fragments with ds_load_b64.
// A-frag (16x4 f32): lanes 0-15 -> K=kc,kc+1 ; lanes 16-31 -> K=kc+2,kc+3.
// C/D-frag: VGPR r -> M = r (lanes 0-15) / r+8 (lanes 16-31), N = lane&15.
// ---------------------------------------------------------------------------
__global__ __launch_bounds__(32)
void conv2d_wmma(const float* __restrict__ X, const float* __restrict__ W,
                 float* __restrict__ Y,
                 int B, int Cin, int H, int Win,
                 int Cout, int OH, int OW, int stride, int pad) {
    __shared__ float As[16 * LDS_STRIDE];   // [m][k_local]
    __shared__ float Bs[16 * LDS_STRIDE];   // [n][k_local]

    const int lane = threadIdx.x;
    const int half = lane & 15;
    const int hi   = lane >> 4;
    const int K    = Cin * 9;

    const int mBase = blockIdx.y * 16;
    const int nBase = blockIdx.x * 16;

    // this lane's output column (also its staging column for Bs)
    const int ncol = nBase + half;
    const int bImg = ncol / (OH * OW);
    const int rem  = ncol % (OH * OW);
    const int oh = rem / OW, ow = rem % OW;
    const int ih0 = oh * stride - pad;
    const int iw0 = ow * stride - pad;
    const float* Xb = X + (size_t)bImg * Cin * H * Win;
    const float* Wm = W + (size_t)(mBase + half) * K;   // staging row for As

    v8f acc = {0.f, 0.f, 0.f, 0.f, 0.f, 0.f, 0.f, 0.f};

    for (int kc = 0; kc < K; kc += KCHUNK) {
        __builtin_prefetch(Wm + kc + KCHUNK, 0, 1);   // global_prefetch_b8

        // ---- stage A panel: row = half, k_local = hi*16 + j (branchless) ----
#pragma unroll
        for (int j = 0; j < 16; ++j) {
            const int kl = hi * 16 + j;
            const int k  = kc + kl;
            const int idx = (k < K) ? k : (K - 1);     // clamp, load, select
            const float v = Wm[idx];
            As[half * LDS_STRIDE + kl] = (k < K) ? v : 0.f;
        }
        // ---- stage B panel: col = half, k_local = hi*16 + j (branchless) ----
#pragma unroll
        for (int j = 0; j < 16; ++j) {
            const int kl = hi * 16 + j;
            const int k  = kc + kl;
            const int ci = k / 9, p = k % 9;
            const int ih = ih0 + p / 3;
            const int iw = iw0 + p % 3;
            const bool inb = (k < K) && (ih >= 0) && (ih < H) && (iw >= 0) && (iw < Win);
            const int off = inb ? ((ci * H + ih) * Win + iw) : 0;
            const float v = Xb[off];
            Bs[half * LDS_STRIDE + kl] = inb ? v : 0.f;
        }
        __syncthreads();

        // ---- 8 WMMAs over the staged 32-deep chunk ----
#pragma unroll
        for (int ks = 0; ks < KCHUNK; ks += 4) {
            const int k0 = ks + 2 * hi;                // even -> b64 aligned
            const v2f a   = *(const v2f*)&As[half * LDS_STRIDE + k0];
            const v2f bfr = *(const v2f*)&Bs[half * LDS_STRIDE + k0];
            acc = __builtin_amdgcn_wmma_f32_16x16x4_f32(
                false, a, false, bfr, (short)0, acc, false, false);
        }
        __syncthreads();
    }

#pragma unroll
    for (int r = 0; r < 8; ++r) {
        const int m = mBase + r + 8 * hi;   // tile counts divide exactly
        Y[(((size_t)bImg * Cout + m) * OH + oh) * OW + ow] = acc[r];
    }
}

// ---------------------------------------------------------------------------
// BatchNorm2d (training mode) batch statistics: one workgroup per channel.
// mv[c] = mean, mv[C+c] = biased variance.
// ---------------------------------------------------------------------------
__global__ __launch_bounds__(256)
void bn_stats(const float* __restrict__ X, float* __restrict__ mv,
              int B, int C, int HW) {
    __shared__ float s1[256], s2[256];
    const int c = blockIdx.x;
    const int cnt = B * HW;
    float sum = 0.f, sq = 0.f;
    for (int idx = threadIdx.x; idx < cnt; idx += 256) {
        const int b = idx / HW, s = idx % HW;
        const float v = X[((size_t)b * C + c) * HW + s];
        sum += v; sq += v * v;
    }
    s1[threadIdx.x] = sum; s2[threadIdx.x] = sq;
    __syncthreads();
    for (int off = 128; off > 0; off >>= 1) {
        if (threadIdx.x < off) {
            s1[threadIdx.x] += s1[threadIdx.x + off];
            s2[threadIdx.x] += s2[threadIdx.x + off];
        }
        __syncthreads();
    }
    if (threadIdx.x == 0) {
        const float m = s1[0] / (float)cnt;
        mv[c]     = m;
        mv[C + c] = s2[0] / (float)cnt - m * m;
    }
}

__global__ __launch_bounds__(256)
void bn_apply_relu(float* __restrict__ X, const float* __restrict__ mv,
                   const float* __restrict__ g, const float* __restrict__ b,
                   int C, int HW, int total) {
    const int i = blockIdx.x * 256 + threadIdx.x;
    if (i >= total) return;
    const int c = (i / HW) % C;
    const float y = (X[i] - mv[c]) * rsqrtf(mv[C + c] + EPS_BN) * g[c] + b[c];
    X[i] = y > 0.f ? y : 0.f;
}

// ---------------------------------------------------------------------------
// Primary caps: add channel bias, squash over 12-long groups of the FLAT
// (B,192,7,7) buffer (faithful to the torch .view(B,784,12)).
// ---------------------------------------------------------------------------
__global__ __launch_bounds__(256)
void primary_squash(const float* __restrict__ pc, const float* __restrict__ bp,
                    float* __restrict__ u, int total_caps) {
    const int cap = blockIdx.x * 256 + threadIdx.x;
    if (cap >= total_caps) return;
    const int b = cap / 784, g = cap % 784;
    float vec[12];
    float ss = 0.f;
#pragma unroll
    for (int i = 0; i < 12; ++i) {
        const int f = g * 12 + i;          // flat within 9408 = 192*49
        const int c = f / 49;              // channel for bias
        const float x = pc[(size_t)b * 9408 + f] + bp[c];
        vec[i] = x; ss += x * x;
    }
    const float scale = ss / ((1.f + ss) * sqrtf(ss + EPS_SQ));
#pragma unroll
    for (int i = 0; i < 12; ++i) u[(size_t)cap * 12 + i] = scale * vec[i];
}

// ---------------------------------------------------------------------------
// Fused ConvCaps layer + dynamic routing. One workgroup per (b,t).
// u_hat (144x16x12 = 110 KB) is held entirely in CDNA5's 320 KB LDS, so all
// three routing iterations never touch HBM.
// ---------------------------------------------------------------------------
__global__ __launch_bounds__(256)
void convcaps_route(const float* __restrict__ u, const float* __restrict__ w_cc,
                    float* __restrict__ u2) {
    extern __shared__ float sm[];
    float* uu   = sm;                 // 1728   : uu[k,i]
    float* uh   = uu + 1728;          // 27648  : u_hat[k, cap*12+d]
    float* blog = uh + 27648;         // 2304   : b_log[k,cap]
    float* cc   = blog + 2304;        // 2304   : routing coefficients
    float* sbuf = cc + 2304;          // 192
    float* vbuf = sbuf + 192;         // 192
    const int tid = threadIdx.x;
    const int bt = blockIdx.x;
    const int b = bt / 49, t = bt % 49;

    // gather uu tile: faithful torch reinterpretation of the unfold buffer
    for (int f = tid; f < 1728; f += 256) {
        const int gf = t * 1728 + f;
        const int cp = gf / 49, L = gf % 49;
        const int c = cp / 9, p = cp % 9;
        const int ih = L / 7 + p / 3 - 1;
        const int iw = L % 7 + p % 3 - 1;
        const bool inb = (ih >= 0) && (ih < 7) && (iw >= 0) && (iw < 7);
        const int off = inb ? ((((b * 192 + c) * 7 + ih) * 7) + iw) : 0;
        const float v = u[off];
        uu[f] = inb ? v : 0.f;
    }
    for (int f = tid; f < 2304; f += 256) blog[f] = 0.f;
    __syncthreads();

    // u_hat[k,o] = w_cc[k,o,:] . uu[k,:]
    for (int idx = tid; idx < 144 * 192; idx += 256) {
        const int k = idx / 192, o = idx % 192;
        const float* w  = w_cc + ((size_t)k * 192 + o) * 12;
        const float* uv = uu + k * 12;
        float s = 0.f;
#pragma unroll
        for (int i = 0; i < 12; ++i) s += w[i] * uv[i];
        uh[idx] = s;
    }
    __syncthreads();

    for (int r = 0; r < 3; ++r) {
        // softmax over the 16 out-caps for each k
        for (int k = tid; k < 144; k += 256) {
            float mx = blog[k * 16];
            for (int o = 1; o < 16; ++o) mx = fmaxf(mx, blog[k * 16 + o]);
            float den = 0.f;
            for (int o = 0; o < 16; ++o) {
                const float e = __expf(blog[k * 16 + o] - mx);
                cc[k * 16 + o] = e; den += e;
            }
            const float inv = 1.f / den;
            for (int o = 0; o < 16; ++o) cc[k * 16 + o] *= inv;
        }
        __syncthreads();
        // s[cap,d] = sum_k c[k,cap] * u_hat[k,cap,d]
        if (tid < 192) {
            const int o = tid / 12, d = tid % 12;
            float s = 0.f;
            for (int k = 0; k < 144; ++k)
                s += cc[k * 16 + o] * uh[k * 192 + o * 12 + d];
            sbuf[tid] = s;
        }
        __syncthreads();
        // squash each of the 16 capsules
        if (tid < 16) {
            float ss = 0.f;
            for (int d = 0; d < 12; ++d) { const float x = sbuf[tid * 12 + d]; ss += x * x; }
            const float scale = ss / ((1.f + ss) * sqrtf(ss + EPS_SQ));
            for (int d = 0; d < 12; ++d) vbuf[tid * 12 + d] = scale * sbuf[tid * 12 + d];
        }
        __syncthreads();
        if (r < 2) {
            for (int ko = tid; ko < 2304; ko += 256) {
                const int k = ko / 16, o = ko % 16;
                float s = 0.f;
#pragma unroll
                for (int d = 0; d < 12; ++d)
                    s += vbuf[o * 12 + d] * uh[k * 192 + o * 12 + d];
                blog[ko] += s;
            }
            __syncthreads();
        }
    }
    if (tid < 192)
        u2[((size_t)b * 784 + t * 16 + tid / 12) * 12 + tid % 12] = vbuf[tid];
}

// ---------------------------------------------------------------------------
// Class caps u_hat2 = einsum('nodi,bni->bnod') as per-n GEMM (64x12)@(12x160)
// via V_WMMA_F32_16X16X4_F32. grid = (n=784, od-tile=10, b-tile=4), one wave.
// K=12 -> 3 straight-line WMMAs, all loads in range (k0 <= 11).
// ---------------------------------------------------------------------------
__global__ __launch_bounds__(32)
void class_uhat_wmma(const float* __restrict__ u2, const float* __restrict__ w_class,
                     float* __restrict__ uhat2) {
    const int lane = threadIdx.x;
    const int half = lane & 15;
    const int hi   = lane >> 4;
    const int n = blockIdx.x;                // 0..783
    const int odBase = blockIdx.y * 16;      // 10 tiles of 16 over o*16+d
    const int bBase  = blockIdx.z * 16;      // 4 tiles over batch

    const int odCol = odBase + half;
    const float* Arow = u2 + ((size_t)(bBase + half) * 784 + n) * 12;
    const float* Bcol = w_class + ((size_t)n * 160 + odCol) * 12; // ((n*10+o)*16+d)*12

    v8f acc = {0.f, 0.f, 0.f, 0.f, 0.f, 0.f, 0.f, 0.f};
#pragma unroll
    for (int kc = 0; kc < 12; kc += 4) {
        const int k0 = kc + 2 * hi;          // max 11: always in range
        v2f a, bb;
        a.x  = Arow[k0];  a.y  = Arow[k0 + 1];
        bb.x = Bcol[k0];  bb.y = Bcol[k0 + 1];
        acc = __builtin_amdgcn_wmma_f32_16x16x4_f32(
            false, a, false, bb, (short)0, acc, false, false);
    }
#pragma unroll
    for (int r = 0; r < 8; ++r) {
        const int bIdx = bBase + r + 8 * hi;
        uhat2[((size_t)bIdx * 784 + n) * 160 + odCol] = acc[r];
    }
}

// ---------------------------------------------------------------------------
// Class-caps dynamic routing: one workgroup per batch image; b-logits in LDS.
// ---------------------------------------------------------------------------
__global__ __launch_bounds__(256)
void class_route(const float* __restrict__ uhat2, float* __restrict__ out) {
    extern __shared__ float sm[];
    float* blog = sm;             // 7840 (784*10)
    float* cc   = blog + 7840;    // 7840
    float* sbuf = cc + 7840;      // 160
    float* vbuf = sbuf + 160;     // 160
    const int tid = threadIdx.x;
    const int b = blockIdx.x;
    const float* uh = uhat2 + (size_t)b * 784 * 160;

    for (int i = tid; i < 7840; i += 256) blog[i] = 0.f;
    __syncthreads();

    for (int r = 0; r < 3; ++r) {
        for (int n = tid; n < 784; n += 256) {
            float mx = blog[n * 10];
            for (int o = 1; o < 10; ++o) mx = fmaxf(mx, blog[n * 10 + o]);
            float den = 0.f;
            for (int o = 0; o < 10; ++o) {
                const float e = __expf(blog[n * 10 + o] - mx);
                cc[n * 10 + o] = e; den += e;
            }
            const float inv = 1.f / den;
            for (int o = 0; o < 10; ++o) cc[n * 10 + o] *= inv;
        }
        __syncthreads();
        if (tid < 160) {
            const int o = tid / 16, d = tid % 16;
            float s = 0.f;
            for (int n = 0; n < 784; ++n)
                s += cc[n * 10 + o] * uh[n * 160 + o * 16 + d];
            sbuf[tid] = s;
        }
        __syncthreads();
        if (tid < 10) {
            float ss = 0.f;
            for (int d = 0; d < 16; ++d) { const float x = sbuf[tid * 16 + d]; ss += x * x; }
            const float scale = ss / ((1.f + ss) * sqrtf(ss + EPS_SQ));
            for (int d = 0; d < 16; ++d) vbuf[tid * 16 + d] = scale * sbuf[tid * 16 + d];
        }
        __syncthreads();
        if (r < 2) {
            for (int no = tid; no < 7840; no += 256) {
                const int n = no / 10, o = no % 10;
                float s = 0.f;
                for (int d = 0; d < 16; ++d)
                    s += vbuf[o * 16 + d] * uh[n * 160 + o * 16 + d];
                blog[no] += s;
            }
            __syncthreads();
        }
    }
    if (tid < 160) out[(size_t)b * 160 + tid] = vbuf[tid];
}

// ---------------------------------------------------------------------------
// Host launcher
// ---------------------------------------------------------------------------
extern "C" void kernel_launch(void* const* d_in, const int* in_sizes, int n_in,
                              void* d_out, int out_size, void* d_ws, size_t ws_size,
                              hipStream_t stream) {
    (void)in_sizes; (void)n_in; (void)out_size; (void)ws_size;
    const float* images  = (const float*)d_in[0];
    const float* w1      = (const float*)d_in[1];
    const float* g1      = (const float*)d_in[2];
    const float* b1      = (const float*)d_in[3];
    const float* w2      = (const float*)d_in[4];
    const float* g2      = (const float*)d_in[5];
    const float* b2      = (const float*)d_in[6];
    const float* w3      = (const float*)d_in[7];
    const float* g3      = (const float*)d_in[8];
    const float* b3      = (const float*)d_in[9];
    const float* wp      = (const float*)d_in[10];
    const float* bp      = (const float*)d_in[11];
    const float* w_cc    = (const float*)d_in[12];
    const float* w_class = (const float*)d_in[13];
    float* out = (float*)d_out;

    float* ws = (float*)d_ws;
    float* x1    = ws;                       // 64*48*32*32  = 3,145,728
    float* x2    = x1 + 3145728;             // 64*96*16*16  = 1,572,864
    float* x3    = x2 + 1572864;             // 64*192*16*16 = 3,145,728
    float* pc    = x3 + 3145728;             // 64*192*7*7   =   602,112
    float* u     = pc + 602112;              //                 602,112
    float* u2    = u  + 602112;              //                 602,112
    float* uhat2 = u2 + 602112;              // 64*784*160   = 8,028,160
    float* mv    = uhat2 + 8028160;          // 2*192 (mean|var)

    // conv1: (64,3,32,32) -> (64,48,32,32), s=1 p=1 ; BN + ReLU
    conv2d_wmma<<<dim3(4096, 3), 32, 0, stream>>>(images, w1, x1,
                                                  64, 3, 32, 32, 48, 32, 32, 1, 1);
    bn_stats<<<48, 256, 0, stream>>>(x1, mv, 64, 48, 1024);
    bn_apply_relu<<<12288, 256, 0, stream>>>(x1, mv, g1, b1, 48, 1024, 3145728);

    // conv2: -> (64,96,16,16), s=2 p=1 ; BN + ReLU
    conv2d_wmma<<<dim3(1024, 6), 32, 0, stream>>>(x1, w2, x2,
                                                  64, 48, 32, 32, 96, 16, 16, 2, 1);
    bn_stats<<<96, 256, 0, stream>>>(x2, mv, 64, 96, 256);
    bn_apply_relu<<<6144, 256, 0, stream>>>(x2, mv, g2, b2, 96, 256, 1572864);

    // conv3: -> (64,192,16,16), s=1 p=1 ; BN + ReLU
    conv2d_wmma<<<dim3(1024, 12), 32, 0, stream>>>(x2, w3, x3,
                                                   64, 96, 16, 16, 192, 16, 16, 1, 1);
    bn_stats<<<192, 256, 0, stream>>>(x3, mv, 64, 192, 256);
    bn_apply_relu<<<12288, 256, 0, stream>>>(x3, mv, g3, b3, 192, 256, 3145728);

    // primary caps conv: -> (64,192,7,7), s=2 p=0 ; + bias ; squash
    conv2d_wmma<<<dim3(196, 12), 32, 0, stream>>>(x3, wp, pc,
                                                  64, 192, 16, 16, 192, 7, 7, 2, 0);
    primary_squash<<<196, 256, 0, stream>>>(pc, bp, u, 64 * 784);

    // ConvCaps layer + routing, fused per (b,t) with u_hat in LDS
    const size_t cc_shmem = (size_t)(1728 + 27648 + 2304 + 2304 + 192 + 192) * sizeof(float);
    convcaps_route<<<64 * 49, 256, cc_shmem, stream>>>(u, w_cc, u2);

    // Class caps u_hat2 via WMMA, then routing per batch image
    class_uhat_wmma<<<dim3(784, 10, 4), 32, 0, stream>>>(u2, w_class, uhat2);
    const size_t cr_shmem = (size_t)(7840 + 7840 + 160 + 160) * sizeof(float);
    class_route<<<64, 256, cr_shmem, stream>>>(uhat2, out);
}